// TrojanDetector_15247133901527
// MI455X (gfx1250) — compile-verified
//
#include <hip/hip_runtime.h>
#include <hip/hip_bf16.h>

#define N_NODES   50000
#define N_EDGES   800000
#define CH        128
#define NUM_GRAPHS 64
#define OUT_CH    2

typedef float v2f __attribute__((ext_vector_type(2)));
typedef float v8f __attribute__((ext_vector_type(8)));

// ---------------------------------------------------------------- utilities
__global__ void fill_kernel(float* __restrict__ p, float v, int n) {
    int i = blockIdx.x * blockDim.x + threadIdx.x;
    if (i < n) p[i] = v;
}

// deg[dst] += 1 for every edge
__global__ void degree_kernel(const int* __restrict__ dst, float* __restrict__ deg, int n) {
    int e = blockIdx.x * blockDim.x + threadIdx.x;
    if (e < n) atomicAdd(&deg[dst[e]], 1.0f);
}

// dinv[i] = (deg[i] + 1)^-0.5   (in place)
__global__ void dinv_kernel(float* __restrict__ dinv, int n) {
    int i = blockIdx.x * blockDim.x + threadIdx.x;
    if (i < n) dinv[i] = rsqrtf(dinv[i] + 1.0f);
}

// ---------------------------------------------------------------- WMMA GEMM
// out[N,128] = H[N,128] @ W[128,128]  (all f32, row-major)
// block = 128 threads (4 waves). Each block: 16 rows x 128 cols.
// Wave w owns cols [32w, 32w+32) as two 16x16 f32 accumulator tiles.
// K-loop: 32 steps of V_WMMA_F32_16X16X4_F32.
__global__ __launch_bounds__(128)
void gemm_wmma_kernel(const float* __restrict__ H, const float* __restrict__ W,
                      float* __restrict__ out) {
    // +4 padding: column walk (stride 132) hits 16 distinct LDS banks per half-wave
    __shared__ float Hs[16][132];

    const int row0 = blockIdx.x * 16;
    const int tid  = threadIdx.x;

    // cooperative load of the 16x128 H tile
    for (int i = tid; i < 16 * CH; i += 128) {
        int r = i >> 7, c = i & 127;
        Hs[r][c] = H[(row0 + r) * CH + c];
    }
    __syncthreads();

    const int lane  = tid & 31;
    const int wave  = tid >> 5;
    const int m     = lane & 15;          // row/col within fragment
    const int khalf = (lane >> 4) * 2;    // lanes 16..31 carry K+2, K+3
    const int n0    = wave * 32;

    v8f c0 = {}; v8f c1 = {};

    for (int k = 0; k < CH; k += 4) {
        // A fragment: 16x4 tile of H rows row0..row0+15
        v2f a;
        a.x = Hs[m][k + khalf];
        a.y = Hs[m][k + khalf + 1];

        // B fragments: 4x16 tiles of W (row-major [K][N]); coalesced 64B per half-wave
        const float* Wr0 = W + (k + khalf) * CH;
        const float* Wr1 = W + (k + khalf + 1) * CH;
        v2f b0, b1;
        b0.x = Wr0[n0 + m];       b0.y = Wr1[n0 + m];
        b1.x = Wr0[n0 + 16 + m];  b1.y = Wr1[n0 + 16 + m];

        c0 = __builtin_amdgcn_wmma_f32_16x16x4_f32(false, a, false, b0,
                                                   (short)0, c0, false, false);
        c1 = __builtin_amdgcn_wmma_f32_16x16x4_f32(false, a, false, b1,
                                                   (short)0, c1, false, false);
    }

    // C/D layout: VGPR r -> M=r (lanes 0-15), M=r+8 (lanes 16-31); N = lane&15
    const int rbase = (lane >> 4) * 8;
    for (int r = 0; r < 8; ++r) {
        int row = row0 + rbase + r;
        out[row * CH + n0 + m]      = c0[r];
        out[row * CH + n0 + 16 + m] = c1[r];
    }
}

// ---------------------------------------------------------------- edge scatter
// agg[dst] += dinv[src]*dinv[dst] * hw[src]; 4 channels per thread (float4 gather)
__global__ void scatter_kernel(const int* __restrict__ src, const int* __restrict__ dst,
                               const float* __restrict__ dinv, const float* __restrict__ hw,
                               float* __restrict__ agg) {
    int idx = blockIdx.x * blockDim.x + threadIdx.x;   // N_EDGES * 32 threads
    int e   = idx >> 5;
    if (e >= N_EDGES) return;
    int c4  = (idx & 31) * 4;
    int s = src[e], d = dst[e];
    float ne = dinv[s] * dinv[d];
    const float4 v = *reinterpret_cast<const float4*>(hw + s * CH + c4);
    float* p = agg + d * CH + c4;
    atomicAdd(p + 0, ne * v.x);
    atomicAdd(p + 1, ne * v.y);
    atomicAdd(p + 2, ne * v.z);
    atomicAdd(p + 3, ne * v.w);
}

// agg = relu(agg + dinv^2 * hw + b)   (in place; agg becomes next layer input)
__global__ void combine_kernel(float* __restrict__ agg, const float* __restrict__ hw,
                               const float* __restrict__ dinv, const float* __restrict__ b) {
    int idx = blockIdx.x * blockDim.x + threadIdx.x;
    if (idx >= N_NODES * CH) return;
    int i = idx >> 7, c = idx & 127;
    float dn = dinv[i];
    float v  = agg[idx] + dn * dn * hw[idx] + b[c];
    agg[idx] = fmaxf(v, 0.0f);
}

// ---------------------------------------------------------------- pooling
__global__ void count_kernel(const int* __restrict__ batch, float* __restrict__ counts, int n) {
    int i = blockIdx.x * blockDim.x + threadIdx.x;
    if (i < n) atomicAdd(&counts[batch[i]], 1.0f);
}

__global__ void pool_kernel(const float* __restrict__ h, const int* __restrict__ batch,
                            float* __restrict__ pooled) {
    int idx = blockIdx.x * blockDim.x + threadIdx.x;
    if (idx >= N_NODES * CH) return;
    int i = idx >> 7, c = idx & 127;
    atomicAdd(&pooled[batch[i] * CH + c], h[idx]);
}

// out[g,o] = sum_c pooled[g,c]/max(counts[g],1) * Wc[c,o] + bc[o]
__global__ void classify_kernel(const float* __restrict__ pooled, const float* __restrict__ counts,
                                const float* __restrict__ Wc, const float* __restrict__ bc,
                                float* __restrict__ out) {
    int t = threadIdx.x;
    if (t >= NUM_GRAPHS * OUT_CH) return;
    int g = t / OUT_CH, o = t % OUT_CH;
    float inv = 1.0f / fmaxf(counts[g], 1.0f);
    float s = 0.0f;
    for (int c = 0; c < CH; ++c)
        s += pooled[g * CH + c] * inv * Wc[c * OUT_CH + o];
    out[t] = s + bc[o];
}

// ---------------------------------------------------------------- launcher
extern "C" void kernel_launch(void* const* d_in, const int* in_sizes, int n_in,
                              void* d_out, int out_size, void* d_ws, size_t ws_size,
                              hipStream_t stream) {
    const float* x     = (const float*)d_in[0];
    const int*   ei    = (const int*)  d_in[1];
    const int*   batch = (const int*)  d_in[2];
    const float* W1    = (const float*)d_in[3];
    const float* b1    = (const float*)d_in[4];
    const float* W2    = (const float*)d_in[5];
    const float* b2    = (const float*)d_in[6];
    const float* W3    = (const float*)d_in[7];
    const float* b3    = (const float*)d_in[8];
    const float* Wc    = (const float*)d_in[9];
    const float* bc    = (const float*)d_in[10];

    const int* src = ei;            // edge_index[0,:]
    const int* dst = ei + N_EDGES;  // edge_index[1,:]

    // workspace carve-out (floats); all offsets 16B-aligned
    float* ws     = (float*)d_ws;
    float* dinv   = ws;                       // 50000   (also used as deg)
    float* hw     = ws + 50048;               // 6.4M
    float* agg    = hw + (size_t)N_NODES * CH; // 6.4M  (doubles as h_in/h_out)
    float* pooled = agg + (size_t)N_NODES * CH; // 8192
    float* counts = pooled + NUM_GRAPHS * CH;   // 64

    const int B = 256;
    auto blocks = [](int n, int b) { return (n + b - 1) / b; };

    // degrees -> dinv
    fill_kernel<<<blocks(N_NODES, B), B, 0, stream>>>(dinv, 0.0f, N_NODES);
    degree_kernel<<<blocks(N_EDGES, B), B, 0, stream>>>(dst, dinv, N_EDGES);
    dinv_kernel<<<blocks(N_NODES, B), B, 0, stream>>>(dinv, N_NODES);

    const float* Ws[3] = {W1, W2, W3};
    const float* bs[3] = {b1, b2, b3};
    const float* h = x;
    for (int l = 0; l < 3; ++l) {
        gemm_wmma_kernel<<<N_NODES / 16, 128, 0, stream>>>(h, Ws[l], hw);
        fill_kernel<<<blocks(N_NODES * CH, B), B, 0, stream>>>(agg, 0.0f, N_NODES * CH);
        scatter_kernel<<<blocks(N_EDGES * 32, B), B, 0, stream>>>(src, dst, dinv, hw, agg);
        combine_kernel<<<blocks(N_NODES * CH, B), B, 0, stream>>>(agg, hw, dinv, bs[l]);
        h = agg;
    }

    // mean pool + classifier
    fill_kernel<<<blocks(NUM_GRAPHS * CH + NUM_GRAPHS, B), B, 0, stream>>>(
        pooled, 0.0f, NUM_GRAPHS * CH + NUM_GRAPHS);
    count_kernel<<<blocks(N_NODES, B), B, 0, stream>>>(batch, counts, N_NODES);
    pool_kernel<<<blocks(N_NODES * CH, B), B, 0, stream>>>(agg, batch, pooled);
    classify_kernel<<<1, 128, 0, stream>>>(pooled, counts, Wc, bc, (float*)d_out);
}